// GATv2_85753317032575
// MI455X (gfx1250) — compile-verified
//
#include <hip/hip_runtime.h>
#include <math.h>

// ---------------------------------------------------------------------------
// Problem constants (from the reference)
// ---------------------------------------------------------------------------
namespace {
constexpr int  Nn   = 30000;
constexpr int  Ee   = 480000;
constexpr int  Fin  = 128;
constexpr int  Dd   = 32;
constexpr int  Hh   = 4;
constexpr int  Gg   = 128;
constexpr int  Tt   = 10;
constexpr int  HD   = Hh * Dd;          // 128
constexpr long EP   = (long)Ee + Nn;    // edges incl. self loops = 510000
constexpr float SLOPE = 0.2f;
}

typedef __attribute__((ext_vector_type(2))) float v2f;
typedef __attribute__((ext_vector_type(8))) float v8f;

// ---------------------------------------------------------------------------
// Helpers
// ---------------------------------------------------------------------------
__device__ __forceinline__ void atomicMaxF(float* addr, float val) {
  // valid for buffers initialized to -inf
  if (val >= 0.0f) atomicMax((int*)addr, __float_as_int(val));
  else             atomicMin((unsigned int*)addr, __float_as_uint(val));
}

__device__ __forceinline__ void edgeSD(const int* __restrict__ ei, long e,
                                       int& s, int& d) {
  if (e < (long)Ee) { s = ei[e]; d = ei[(long)Ee + e]; }
  else              { s = d = (int)(e - Ee); }
}

// ---------------------------------------------------------------------------
// WMMA GEMM:  C[M x Nout] = A[M x K] @ W[Nout x K]^T + bias[Nout]
// f32 WMMA 16x16x4, one 16x16 C tile per wave. M%16==0, Nout%16==0, K%4==0.
// A-matrix layout (ISA 7.12.2, 32-bit A 16x4): lanes 0-15 -> K=k0,k0+1 in
// VGPR0/1; lanes 16-31 -> K=k0+2,k0+3.  B fed symmetrically from W^T.
// ---------------------------------------------------------------------------
__global__ void gemm_wmma_bias(const float* __restrict__ A,
                               const float* __restrict__ W,
                               const float* __restrict__ bias,
                               float* __restrict__ C,
                               int M, int Nout, int K) {
  const int lane  = threadIdx.x & 31;
  const int wave  = blockIdx.x * (blockDim.x >> 5) + (threadIdx.x >> 5);
  const int ntile = Nout >> 4;
  const int mtile = M >> 4;
  const int mt    = wave / ntile;
  const int nt    = wave - mt * ntile;
  if (mt >= mtile) return;              // wave-uniform: EXEC all-1 below

  const int  half = lane >> 4;          // 0: K=k0,k0+1   1: K=k0+2,k0+3
  const int  l15  = lane & 15;
  const long arow = (long)(mt * 16 + l15) * K;
  const long wrow = (long)(nt * 16 + l15) * K;

  v8f acc = {0.f, 0.f, 0.f, 0.f, 0.f, 0.f, 0.f, 0.f};
  for (int k0 = 0; k0 < K; k0 += 4) {
    const int ks = k0 + half * 2;
    v2f a = *(const v2f*)(A + arow + ks);   // A[row][ks], A[row][ks+1]
    v2f b = *(const v2f*)(W + wrow + ks);   // B[k][col] = W[col][k]
    acc = __builtin_amdgcn_wmma_f32_16x16x4_f32(
        /*neg_a=*/false, a, /*neg_b=*/false, b,
        /*c_mod=*/(short)0, acc, /*reuse_a=*/false, /*reuse_b=*/false);
  }

  // C/D layout: VGPR i -> row mt*16 + i + half*8, col nt*16 + l15
  const int   col   = nt * 16 + l15;
  const float bb    = bias[col];
  const int   rbase = mt * 16 + half * 8;
#pragma unroll
  for (int i = 0; i < 8; ++i)
    C[(long)(rbase + i) * Nout + col] = acc[i] + bb;
}

// ---------------------------------------------------------------------------
// Fill buffer with a value (used for -inf init of segment-max buffers)
// ---------------------------------------------------------------------------
__global__ void fill_val(float* __restrict__ p, long n, float v) {
  long t = (long)blockIdx.x * blockDim.x + threadIdx.x;
  if (t < n) p[t] = v;
}

// ---------------------------------------------------------------------------
// Edge phase 1: alpha[e,h] = sum_d leaky(xl[src]+xr[dst]) * att ;
//               segment max into amax[dst,h]; in-degree count (h==0)
// ---------------------------------------------------------------------------
__global__ void edge_alpha(const int* __restrict__ ei,
                           const float* __restrict__ xl,
                           const float* __restrict__ xr,
                           const float* __restrict__ att,
                           float* __restrict__ exb,
                           float* __restrict__ amax,
                           float* __restrict__ cnt) {
  long t = (long)blockIdx.x * blockDim.x + threadIdx.x;
  if (t >= EP * Hh) return;
  const int  h = (int)(t & (Hh - 1));
  const long e = t >> 2;                       // Hh == 4
  int s, d; edgeSD(ei, e, s, d);

  const float* pl = xl + (long)s * HD + h * Dd;
  const float* pr = xr + (long)d * HD + h * Dd;
  const float* pa = att + h * Dd;
  float acc = 0.0f;
#pragma unroll 4
  for (int k = 0; k < Dd; ++k) {
    float v = pl[k] + pr[k];
    v = (v > 0.0f) ? v : v * SLOPE;
    acc += v * pa[k];
  }
  exb[t] = acc;
  atomicMaxF(&amax[(long)d * Hh + h], acc);
  if (h == 0) atomicAdd(&cnt[d], 1.0f);
}

// ---------------------------------------------------------------------------
// Edge phase 2: ex = exp(alpha - amax[dst]); denom[dst,h] += ex
// ---------------------------------------------------------------------------
__global__ void edge_exp(const int* __restrict__ ei,
                         float* __restrict__ exb,
                         const float* __restrict__ amax,
                         float* __restrict__ denom) {
  long t = (long)blockIdx.x * blockDim.x + threadIdx.x;
  if (t >= EP * Hh) return;
  const int  h = (int)(t & (Hh - 1));
  const long e = t >> 2;
  int s, d; edgeSD(ei, e, s, d);
  float ex = expf(exb[t] - amax[(long)d * Hh + h]);
  exb[t] = ex;
  atomicAdd(&denom[(long)d * Hh + h], ex);
}

// ---------------------------------------------------------------------------
// Edge phase 3: s[dst, j] += xl[src, j] * (ex / denom[dst, h])
// one thread per (edge, channel j in [0,HD))
// ---------------------------------------------------------------------------
__global__ void edge_scatter(const int* __restrict__ ei,
                             const float* __restrict__ xl,
                             const float* __restrict__ exb,
                             const float* __restrict__ denom,
                             float* __restrict__ sbuf) {
  long t = (long)blockIdx.x * blockDim.x + threadIdx.x;
  if (t >= EP * HD) return;
  const int  j = (int)(t & (HD - 1));
  const long e = t >> 7;                       // HD == 128
  const int  h = j >> 5;                       // Dd == 32
  int s, d; edgeSD(ei, e, s, d);
  float a = exb[e * Hh + h] / denom[(long)d * Hh + h];
  atomicAdd(&sbuf[(long)d * HD + j], xl[(long)s * HD + j] * a);
}

// ---------------------------------------------------------------------------
// Node finish: s = s / cnt + conv_bias  (mean aggregation + bias)
// ---------------------------------------------------------------------------
__global__ void node_finish(float* __restrict__ sbuf,
                            const float* __restrict__ cnt,
                            const float* __restrict__ cb) {
  long t = (long)blockIdx.x * blockDim.x + threadIdx.x;
  if (t >= (long)Nn * HD) return;
  const int n = (int)(t >> 7);
  const int j = (int)(t & (HD - 1));
  sbuf[t] = sbuf[t] / cnt[n] + cb[j];
}

// ---------------------------------------------------------------------------
// Global max pool over batch: pooled[batch[n], d] = max(..., h[n, d])
// ---------------------------------------------------------------------------
__global__ void pool_max(const float* __restrict__ hbuf,
                         const int* __restrict__ batch,
                         float* __restrict__ pooled) {
  long t = (long)blockIdx.x * blockDim.x + threadIdx.x;
  if (t >= (long)Nn * Dd) return;
  const int n = (int)(t >> 5);
  const int d = (int)(t & (Dd - 1));
  atomicMaxF(&pooled[(long)batch[n] * Dd + d], hbuf[t]);
}

// ---------------------------------------------------------------------------
// fc1: relu(pooled @ fc1W^T + b)    [G x D]
// ---------------------------------------------------------------------------
__global__ void fc1_relu(const float* __restrict__ pooled,
                         const float* __restrict__ Wm,
                         const float* __restrict__ b,
                         float* __restrict__ out) {
  int t = blockIdx.x * blockDim.x + threadIdx.x;
  if (t >= Gg * Dd) return;
  const int g = t / Dd, o = t - g * Dd;
  float acc = b[o];
  for (int k = 0; k < Dd; ++k) acc += pooled[g * Dd + k] * Wm[o * Dd + k];
  out[t] = fmaxf(acc, 0.0f);
}

// ---------------------------------------------------------------------------
// fc2: g @ fc2W^T + b    [G x T]
// ---------------------------------------------------------------------------
__global__ void fc2_lin(const float* __restrict__ gin,
                        const float* __restrict__ Wm,
                        const float* __restrict__ b,
                        float* __restrict__ out) {
  int t = blockIdx.x * blockDim.x + threadIdx.x;
  if (t >= Gg * Tt) return;
  const int g = t / Tt, o = t - g * Tt;
  float acc = b[o];
  for (int k = 0; k < Dd; ++k) acc += gin[g * Dd + k] * Wm[o * Dd + k];
  out[t] = acc;
}

// ---------------------------------------------------------------------------
// Host side
// ---------------------------------------------------------------------------
static inline void launch_gemm(const float* A, const float* W, const float* bias,
                               float* C, int M, int Nout, int K, hipStream_t st) {
  const int waves = (M / 16) * (Nout / 16);
  const int wpb   = 8;                       // 256 threads = 8 waves
  const int blocks = (waves + wpb - 1) / wpb;
  gemm_wmma_bias<<<blocks, wpb * 32, 0, st>>>(A, W, bias, C, M, Nout, K);
}

static inline int gridFor(long n, int bs) { return (int)((n + bs - 1) / bs); }

extern "C" void kernel_launch(void* const* d_in, const int* in_sizes, int n_in,
                              void* d_out, int out_size, void* d_ws, size_t ws_size,
                              hipStream_t stream) {
  const float* x     = (const float*)d_in[0];
  const int*   ei    = (const int*)  d_in[1];
  const int*   batch = (const int*)  d_in[2];
  const float* Wl[2] = {(const float*)d_in[3],  (const float*)d_in[11]};
  const float* bl[2] = {(const float*)d_in[4],  (const float*)d_in[12]};
  const float* Wr[2] = {(const float*)d_in[5],  (const float*)d_in[13]};
  const float* br[2] = {(const float*)d_in[6],  (const float*)d_in[14]};
  const float* at[2] = {(const float*)d_in[7],  (const float*)d_in[15]};
  const float* cb[2] = {(const float*)d_in[8],  (const float*)d_in[16]};
  const float* lw[2] = {(const float*)d_in[9],  (const float*)d_in[17]};
  const float* lb[2] = {(const float*)d_in[10], (const float*)d_in[18]};
  const float* fc1W = (const float*)d_in[19];
  const float* fc1b = (const float*)d_in[20];
  const float* fc2W = (const float*)d_in[21];
  const float* fc2b = (const float*)d_in[22];
  float* out = (float*)d_out;

  // Workspace layout (floats, all sub-buffers 8B aligned)
  float* ws     = (float*)d_ws;
  float* xl     = ws;                          // N*HD
  float* xr     = xl     + (long)Nn * HD;      // N*HD
  float* exb    = xr     + (long)Nn * HD;      // EP*H
  float* amax   = exb    + EP * Hh;            // N*H
  float* denom  = amax   + (long)Nn * Hh;      // N*H
  float* cntb   = denom  + (long)Nn * Hh;      // N
  float* sbuf   = cntb   + Nn;                 // N*HD
  float* hbuf   = sbuf   + (long)Nn * HD;      // N*D
  float* pooled = hbuf   + (long)Nn * Dd;      // G*D
  float* tmpg   = pooled + (long)Gg * Dd;      // G*D

  const float NEGINF = -__builtin_huge_valf();
  const int BS = 256;

  const float* feat = x;
  int fin = Fin;
  for (int c = 0; c < 2; ++c) {
    // xl = feat @ Wl^T + bl ; xr = feat @ Wr^T + br   (WMMA f32)
    launch_gemm(feat, Wl[c], bl[c], xl, Nn, HD, fin, stream);
    launch_gemm(feat, Wr[c], br[c], xr, Nn, HD, fin, stream);

    // init segment buffers
    hipMemsetAsync(denom, 0, (size_t)Nn * Hh * sizeof(float), stream);
    hipMemsetAsync(cntb,  0, (size_t)Nn * sizeof(float), stream);
    hipMemsetAsync(sbuf,  0, (size_t)Nn * HD * sizeof(float), stream);
    fill_val<<<gridFor((long)Nn * Hh, BS), BS, 0, stream>>>(amax, (long)Nn * Hh, NEGINF);

    // edge phases
    edge_alpha<<<gridFor(EP * Hh, BS), BS, 0, stream>>>(ei, xl, xr, at[c], exb, amax, cntb);
    edge_exp  <<<gridFor(EP * Hh, BS), BS, 0, stream>>>(ei, exb, amax, denom);
    edge_scatter<<<gridFor(EP * HD, BS), BS, 0, stream>>>(ei, xl, exb, denom, sbuf);

    // mean + bias, then HD->D projection (WMMA f32)
    node_finish<<<gridFor((long)Nn * HD, BS), BS, 0, stream>>>(sbuf, cntb, cb[c]);
    launch_gemm(sbuf, lw[c], lb[c], hbuf, Nn, Dd, HD, stream);

    feat = hbuf;
    fin = Dd;
  }

  // global max pool + MLP head
  fill_val<<<gridFor((long)Gg * Dd, BS), BS, 0, stream>>>(pooled, (long)Gg * Dd, NEGINF);
  pool_max<<<gridFor((long)Nn * Dd, BS), BS, 0, stream>>>(hbuf, batch, pooled);
  fc1_relu<<<gridFor((long)Gg * Dd, BS), BS, 0, stream>>>(pooled, fc1W, fc1b, tmpg);
  fc2_lin <<<gridFor((long)Gg * Tt, BS), BS, 0, stream>>>(tmpg, fc2W, fc2b, out);
  (void)in_sizes; (void)n_in; (void)out_size; (void)ws_size;
}